// MyModel_75677323755650
// MI455X (gfx1250) — compile-verified
//
#include <hip/hip_runtime.h>
#include <hip/hip_bf16.h>

typedef __attribute__((ext_vector_type(2))) float v2f;
typedef __attribute__((ext_vector_type(8))) float v8f;

#define BDIM 64
#define SDIM 512
#define HDIM 768
#define BS   (BDIM * SDIM)   // 32768 tokens
#define KQ   (HDIM / 4)      // 192 = K-quarter per accumulator chain

// ---------------------------------------------------------------------------
// Phase 0: pack the 6 projection columns into a [768 x 16] padded weight
// matrix (row-major, zero-padded cols 6..15).
// cols: 0=W_start[:,0] 1=W_start[:,1] 2=W_end[:,0] 3=W_end[:,1] 4=W_m[:H] 5=W_m[H:]
// ---------------------------------------------------------------------------
__global__ void pad_weights_kernel(const float* __restrict__ Ws,
                                   const float* __restrict__ We,
                                   const float* __restrict__ Wm,
                                   float* __restrict__ Wpad) {
    int i = blockIdx.x * blockDim.x + threadIdx.x;
    if (i >= HDIM * 16) return;
    int k = i >> 4;
    int n = i & 15;
    float v = 0.0f;
    if (n < 2)       v = Ws[k * 2 + n];
    else if (n < 4)  v = We[k * 2 + (n - 2)];
    else if (n == 4) v = Wm[k];
    else if (n == 5) v = Wm[HDIM + k];
    Wpad[i] = v;
}

// ---------------------------------------------------------------------------
// Phase 1: per-token projections via V_WMMA_F32_16X16X4_F32.
// One wave computes a 16-token x 16-col C tile. K=768 is split into four
// independent accumulator chains (K-quarters of 192), giving 4-deep MLP so
// global A-loads pipeline instead of serializing on s_wait_loadcnt 0.
// B-fragments come from LDS (weights staged once per block).
// A fragment (16x4 f32): lane L -> row M=L&15, K pair = 2*(L>>4)+{0,1}.
// C tile: lane L, reg r -> M = r + 8*(L>=16), N = L&15.
// ---------------------------------------------------------------------------
__global__ __launch_bounds__(256)
void token_proj_kernel(const float* __restrict__ rep,
                       const int*   __restrict__ mask,
                       const float* __restrict__ Wpad,
                       const float* __restrict__ b_start,
                       const float* __restrict__ b_end,
                       const float* __restrict__ b_m,
                       float* __restrict__ a_out,
                       float* __restrict__ c_out,
                       int*   __restrict__ scand,
                       int*   __restrict__ ecand) {
    __shared__ float ldsW[HDIM * 16];      // 48 KB staged weights
    __shared__ float tileC[8][16][16];     // 8 KB C-tile shuffle space

    // Stage Wpad -> LDS, coalesced float4 copies (3072 float4 / 256 thr = 12)
    for (int i = threadIdx.x; i < (HDIM * 16) / 4; i += 256)
        ((float4*)ldsW)[i] = ((const float4*)Wpad)[i];
    __syncthreads();

    const int wave  = threadIdx.x >> 5;   // 0..7
    const int lane  = threadIdx.x & 31;
    const int m     = lane & 15;
    const int khalf = lane >> 4;          // 0 or 1 -> K offset 0 or 2

    const int tokBase = (blockIdx.x * 8 + wave) * 16;
    const float* arow = rep + (size_t)(tokBase + m) * HDIM + khalf * 2;

    v8f acc[4] = {v8f{}, v8f{}, v8f{}, v8f{}};

    for (int k0 = 0; k0 < KQ; k0 += 4) {
#pragma unroll
        for (int c = 0; c < 4; ++c) {
            const int kk = c * KQ + k0;
            v2f av = *(const v2f*)(arow + kk);     // A[M=m][kk+2*khalf +0/+1]
            const int r = kk + khalf * 2;
            v2f bv;
            bv.x = ldsW[r * 16 + m];               // B[K=r  ][N=m]
            bv.y = ldsW[(r + 1) * 16 + m];         // B[K=r+1][N=m]
            acc[c] = __builtin_amdgcn_wmma_f32_16x16x4_f32(
                /*neg_a=*/false, av, /*neg_b=*/false, bv,
                /*c_mod=*/(short)0, acc[c], /*reuse_a=*/false, /*reuse_b=*/false);
        }
    }

    v8f accS = (acc[0] + acc[1]) + (acc[2] + acc[3]);

    const int mbase = khalf * 8;
#pragma unroll
    for (int r = 0; r < 8; ++r)
        tileC[wave][mbase + r][m] = accS[r];
    __syncthreads();

    // 128 tokens per block -> first 128 threads each finish one token.
    const int t = threadIdx.x;
    if (t < 128) {
        const int w  = t >> 4;
        const int mm = t & 15;
        const int tok = blockIdx.x * 128 + t;
        const float sl0 = tileC[w][mm][0] + b_start[0];
        const float sl1 = tileC[w][mm][1] + b_start[1];
        const float el0 = tileC[w][mm][2] + b_end[0];
        const float el1 = tileC[w][mm][3] + b_end[1];
        const float av  = tileC[w][mm][4] + b_m[0];
        const float cv  = tileC[w][mm][5];
        const int valid = (mask[tok] != 0);
        scand[tok] = valid && (sl0 <= sl1);
        ecand[tok] = valid && (el0 <= el1);
        a_out[tok] = av;
        c_out[tok] = cv;
    }
}

// ---------------------------------------------------------------------------
// Phase 2 (store-bandwidth-bound): one block per (b,s) row; 128 threads x
// float4 cover the 512 e positions. Writes mask (0/1 float) and scores:
// 134 MB of pure coalesced 16B stores (~6 us at 23.3 TB/s).
// ---------------------------------------------------------------------------
__global__ __launch_bounds__(128)
void span_kernel(const float* __restrict__ a,
                 const float* __restrict__ c,
                 const int*   __restrict__ scand,
                 const int*   __restrict__ ecand,
                 float* __restrict__ mask_out,
                 float* __restrict__ scores_out) {
    const int bs = blockIdx.x;        // b*S + s
    const int b  = bs >> 9;           // S = 512
    const int s  = bs & 511;

    const float av = a[bs];
    const int   sc = scand[bs];
    const int   e0 = threadIdx.x * 4;

    const float4 cv = *(const float4*)(c + b * SDIM + e0);
    const int4   ec = *(const int4*)(ecand + b * SDIM + e0);

    float4 sv, mv;
    sv.x = av + cv.x;
    sv.y = av + cv.y;
    sv.z = av + cv.z;
    sv.w = av + cv.w;
    mv.x = (sc && ec.x && (s <= e0 + 0) && (sv.x > 0.0f)) ? 1.0f : 0.0f;
    mv.y = (sc && ec.y && (s <= e0 + 1) && (sv.y > 0.0f)) ? 1.0f : 0.0f;
    mv.z = (sc && ec.z && (s <= e0 + 2) && (sv.z > 0.0f)) ? 1.0f : 0.0f;
    mv.w = (sc && ec.w && (s <= e0 + 3) && (sv.w > 0.0f)) ? 1.0f : 0.0f;

    const size_t off = (size_t)bs * SDIM + e0;
    *(float4*)(mask_out   + off) = mv;
    *(float4*)(scores_out + off) = sv;
}

// ---------------------------------------------------------------------------
// Launch
// inputs: 0=rep 1=mask 2=W_start 3=b_start 4=W_end 5=b_end 6=W_m 7=b_m
// outputs (concat): span_mask [B,S,S] then scores [B,S,S], both as float
// ---------------------------------------------------------------------------
extern "C" void kernel_launch(void* const* d_in, const int* in_sizes, int n_in,
                              void* d_out, int out_size, void* d_ws, size_t ws_size,
                              hipStream_t stream) {
    const float* rep     = (const float*)d_in[0];
    const int*   mask    = (const int*)  d_in[1];
    const float* W_start = (const float*)d_in[2];
    const float* b_start = (const float*)d_in[3];
    const float* W_end   = (const float*)d_in[4];
    const float* b_end   = (const float*)d_in[5];
    const float* W_m     = (const float*)d_in[6];
    const float* b_m     = (const float*)d_in[7];

    // workspace layout (floats/ints)
    float* Wpad  = (float*)d_ws;                 // 768*16 = 12288 floats
    float* a_ws  = Wpad + HDIM * 16;             // BS floats
    float* c_ws  = a_ws + BS;                    // BS floats
    int*   scand = (int*)(c_ws + BS);            // BS ints
    int*   ecand = scand + BS;                   // BS ints

    float* mask_out   = (float*)d_out;                     // [B,S,S]
    float* scores_out = mask_out + (size_t)BS * SDIM;      // [B,S,S]

    // Phase 0: pack weights -> [768 x 16]
    pad_weights_kernel<<<(HDIM * 16 + 255) / 256, 256, 0, stream>>>(
        W_start, W_end, W_m, Wpad);

    // Phase 1: WMMA token projections (256 blocks x 256 thr = 8 waves/blk)
    token_proj_kernel<<<BS / 128, 256, 0, stream>>>(
        rep, mask, Wpad, b_start, b_end, b_m, a_ws, c_ws, scand, ecand);

    // Phase 2: pairwise broadcast + mask, store-bandwidth bound
    span_kernel<<<BS, 128, 0, stream>>>(
        a_ws, c_ws, scand, ecand, mask_out, scores_out);
}